// MedianConvolution_65807488909796
// MI455X (gfx1250) — compile-verified
//
#include <hip/hip_runtime.h>

typedef __attribute__((ext_vector_type(2))) float v2f;
typedef __attribute__((ext_vector_type(8))) float v8f;

#define IN_C  128
#define OUT_C 64
#define KNBR  32

// ---------------------------------------------------------------------------
// Phase 1: h[n,64] = x[n,128] @ W^T, f32 via V_WMMA_F32_16X16X4_F32.
// One wave -> 16 rows (nodes) x 64 cols (out channels) = 4 accumulators.
// A layout (16x4 f32): lanes 0-15 hold K={k,k+1}, lanes 16-31 hold K={k+2,k+3},
// M = lane%16 striped over lanes. B (4x16) mirrors with N = lane%16.
// C/D: VGPR v, lanes 0-15 -> M=v, lanes 16-31 -> M=v+8, N=lane%16.
// ---------------------------------------------------------------------------
__global__ void __launch_bounds__(256) gemm_wmma_f32(
    const float* __restrict__ x, const float* __restrict__ W,
    float* __restrict__ h, int n) {
  const int lane   = threadIdx.x & 31;
  const int gwave  = blockIdx.x * 8 + (threadIdx.x >> 5);
  const int row0   = gwave * 16;
  if (row0 >= n) return;                    // wave-uniform: EXEC stays all-1s

  const int lrow = lane & 15;
  const int hi   = lane >> 4;               // 0 -> K pair {0,1}; 1 -> {2,3}
  const int kh   = hi * 2;

  int arow = row0 + lrow;
  if (arow >= n) arow = n - 1;              // clamp tail rows (select, no branch)
  const float* xrow = x + (size_t)arow * IN_C;
  const float* wb   = W + (size_t)lrow * IN_C + kh;   // col tile base (+t*16*IN_C)

  v8f acc0 = {}, acc1 = {}, acc2 = {}, acc3 = {};

  #pragma unroll 4
  for (int k = 0; k < IN_C; k += 4) {
    v2f a  = *(const v2f*)(xrow + k + kh);
    v2f b0 = *(const v2f*)(wb + k);
    v2f b1 = *(const v2f*)(wb + 16 * IN_C + k);
    v2f b2 = *(const v2f*)(wb + 32 * IN_C + k);
    v2f b3 = *(const v2f*)(wb + 48 * IN_C + k);
    acc0 = __builtin_amdgcn_wmma_f32_16x16x4_f32(false, a, false, b0, (short)0, acc0, false, false);
    acc1 = __builtin_amdgcn_wmma_f32_16x16x4_f32(false, a, false, b1, (short)0, acc1, false, false);
    acc2 = __builtin_amdgcn_wmma_f32_16x16x4_f32(false, a, false, b2, (short)0, acc2, false, false);
    acc3 = __builtin_amdgcn_wmma_f32_16x16x4_f32(false, a, false, b3, (short)0, acc3, false, false);
  }

  #pragma unroll
  for (int v = 0; v < 8; ++v) {
    const int r = row0 + v + hi * 8;
    if (r < n) {
      float* hr = h + (size_t)r * OUT_C + lrow;
      hr[0]  = acc0[v];
      hr[16] = acc1[v];
      hr[32] = acc2[v];
      hr[48] = acc3[v];
    }
  }
}

// ---------------------------------------------------------------------------
// Phase 2: per (node, channel) gather 32 neighbor values of h and take the
// 16th-smallest (sorted index 15 == reference's lower-median).
// Thread layout: 64 consecutive threads = one node -> node id is wave-uniform,
// so nbrs (int64) loads go through the scalar path; h gathers are coalesced
// (32 contiguous floats per neighbor per wave), L2-resident (h = 25.6MB).
// Median: bitonic-sort each half of 16 (2 x 80 CAS), then rank-15 of the
// union = max_i min(a[i], b[15-i])  (first bitonic-merge stage property).
// ---------------------------------------------------------------------------
__global__ void __launch_bounds__(256) gather_median(
    const float* __restrict__ h, const long long* __restrict__ nbrs,
    float* __restrict__ out, int n) {
  const int tid  = blockIdx.x * 256 + threadIdx.x;
  const int node = __builtin_amdgcn_readfirstlane(tid >> 6);  // wave-uniform
  const int ch   = tid & 63;
  if (node >= n) return;                    // wave-uniform exit

  const long long* __restrict__ nb = nbrs + (size_t)node * KNBR;

  float v[KNBR];
  #pragma unroll
  for (int j = 0; j < KNBR; ++j) {
    const long long nj = nb[j];             // scalar load (uniform address)
    v[j] = h[(size_t)nj * OUT_C + ch];      // coalesced vector load
  }

  // Bitonic sort (ascending) of v[0..15] and v[16..31], fully unrolled.
  #pragma unroll
  for (int half = 0; half < 2; ++half) {
    float* a = &v[half * 16];
    #pragma unroll
    for (int k = 2; k <= 16; k <<= 1) {
      #pragma unroll
      for (int j = k >> 1; j > 0; j >>= 1) {
        #pragma unroll
        for (int i = 0; i < 16; ++i) {
          const int l = i ^ j;
          if (l > i) {
            const float x0 = a[i], x1 = a[l];
            const float lo = fminf(x0, x1);
            const float hv = fmaxf(x0, x1);
            const bool up = ((i & k) == 0);
            a[i] = up ? lo : hv;
            a[l] = up ? hv : lo;
          }
        }
      }
    }
  }

  // rank-15 of union of two sorted 16-arrays = max of pairwise mins.
  float med = fminf(v[0], v[31]);
  #pragma unroll
  for (int i = 1; i < 16; ++i)
    med = fmaxf(med, fminf(v[i], v[31 - i]));

  out[tid] = med;
}

// ---------------------------------------------------------------------------
extern "C" void kernel_launch(void* const* d_in, const int* in_sizes, int n_in,
                              void* d_out, int out_size, void* d_ws, size_t ws_size,
                              hipStream_t stream) {
  (void)n_in; (void)out_size; (void)ws_size;
  const float*      x    = (const float*)d_in[0];        // [N,128] f32
  const long long*  nbrs = (const long long*)d_in[1];    // [N,32]  i64
  const float*      W    = (const float*)d_in[2];        // [64,128] f32
  float*            out  = (float*)d_out;                // [N,64]  f32

  const int n = in_sizes[0] / IN_C;
  float* h = (float*)d_ws;                               // [N,64] f32 scratch

  const int waves   = (n + 15) / 16;
  const int blocks1 = (waves + 7) / 8;                   // 8 waves / 256-thread block
  gemm_wmma_f32<<<blocks1, 256, 0, stream>>>(x, W, h, n);

  const long long total  = (long long)n * OUT_C;
  const int       blocks2 = (int)((total + 255) / 256);
  gather_median<<<blocks2, 256, 0, stream>>>(h, nbrs, out, n);
}